// PotentialPredictor_72206990181095
// MI455X (gfx1250) — compile-verified
//
#include <hip/hip_runtime.h>
#include <math.h>

// ---------------- problem constants (match reference) ----------------
constexpr int CCH  = 64;            // channels
constexpr int NPG  = 2049;          // nodes per graph
constexpr int BGR  = 16;            // graphs
constexpr int NTOT = BGR * NPG;     // 32784 nodes
constexpr int ETOT = NTOT * 16;     // 524544 edges
constexpr int FINC = 739;           // input features
constexpr int LNUM = 4;

typedef __attribute__((ext_vector_type(2))) float v2f;
typedef __attribute__((ext_vector_type(8))) float v8f;

__device__ __forceinline__ float gelu_exact(float x) {
    // torch nn.GELU() default: 0.5*x*(1+erf(x/sqrt(2)))
    return 0.5f * x * (1.0f + erff(x * 0.70710678118654752440f));
}

// =====================================================================
// GEMM  D[M,64] = act(A[M,K] @ B[K,64] + bias[64])
// One block = 256 threads = 8 waves; block tile 32(M) x 64(N);
// each wave owns one 16x16 tile via V_WMMA_F32_16X16X4_F32.
// K staged through LDS in slabs of 32; full slabs are branch-free
// (A row index clamped; stores guarded), tail slab zero-padded.
// Fragments for a whole slab are fetched into registers before the
// 8 WMMAs so the LDS loads batch under graduated s_wait_dscnt.
// =====================================================================
template<bool ACT>
__global__ __launch_bounds__(256) void gemm64_wmma(
    const float* __restrict__ A, const float* __restrict__ Bm,
    const float* __restrict__ bias, float* __restrict__ D,
    int M, int K)
{
    __shared__ float lA[32][33];   // stride 33 -> conflict-free column reads
    __shared__ float lB[32][65];

    const int tid   = threadIdx.x;
    const int lane  = tid & 31;
    const int wid   = tid >> 5;
    const int half  = lane >> 4;       // 0: lanes 0-15, 1: lanes 16-31
    const int l16   = lane & 15;
    const int mtile = wid >> 2;        // 0..1
    const int n0w   = (wid & 3) << 4;  // 0,16,32,48
    const int m0    = blockIdx.x * 32;

    // A-tile loader mapping: 32 rows x 32 k, 4 elems/thread
    const int arow  = tid >> 3;        // 0..31
    const int akx   = (tid & 7) << 2;  // 0..28
    // clamp row so the fast path never predicates loads; garbage rows are
    // never stored (epilogue guard).
    const int growc = min(m0 + arow, M - 1);
    // B-tile loader mapping: 32 k x 64 n, 8 elems/thread
    const int bk    = tid >> 3;        // 0..31
    const int bn    = (tid & 7) << 3;  // 0..56

    const int arowf = mtile * 16 + l16;
    v8f acc = {};

    const int kfull = K & ~31;

    // ---------------- full 32-wide slabs: branch-free staging ----------------
    for (int k0 = 0; k0 < kfull; k0 += 32) {
        {
            const float* ap = A + (size_t)growc * K + k0 + akx;
            lA[arow][akx + 0] = ap[0];
            lA[arow][akx + 1] = ap[1];
            lA[arow][akx + 2] = ap[2];
            lA[arow][akx + 3] = ap[3];
        }
        {
            const float4* bp = (const float4*)(Bm + (size_t)(k0 + bk) * 64 + bn);
            const float4 bv0 = bp[0];
            const float4 bv1 = bp[1];
            lB[bk][bn + 0] = bv0.x; lB[bk][bn + 1] = bv0.y;
            lB[bk][bn + 2] = bv0.z; lB[bk][bn + 3] = bv0.w;
            lB[bk][bn + 4] = bv1.x; lB[bk][bn + 5] = bv1.y;
            lB[bk][bn + 6] = bv1.z; lB[bk][bn + 7] = bv1.w;
        }
        __syncthreads();

        // fetch all slab fragments first (batched ds_loads), then compute
        v2f afr[8], bfr[8];
        #pragma unroll
        for (int s = 0; s < 8; ++s) {
            const int kb = (s << 2) + (half << 1);
            // A 16x4: VGPR0 = K{0|2}, VGPR1 = K{1|3} by lane half
            afr[s].x = lA[arowf][kb];
            afr[s].y = lA[arowf][kb + 1];
            // B 4x16: N = lane&15, same K split by lane half
            bfr[s].x = lB[kb][n0w + l16];
            bfr[s].y = lB[kb + 1][n0w + l16];
        }
        #pragma unroll
        for (int s = 0; s < 8; ++s) {
            acc = __builtin_amdgcn_wmma_f32_16x16x4_f32(
                      false, afr[s], false, bfr[s], (short)0, acc, false, false);
        }
        __syncthreads();
    }

    // ---------------- tail slab (only when K % 32 != 0): zero-padded ----------------
    if (kfull < K) {
        const int k0 = kfull;
        {
            float a0 = 0.f, a1 = 0.f, a2 = 0.f, a3 = 0.f;
            const float* ap = A + (size_t)growc * K + k0 + akx;
            const int rem = K - (k0 + akx);
            if (rem > 0) a0 = ap[0];
            if (rem > 1) a1 = ap[1];
            if (rem > 2) a2 = ap[2];
            if (rem > 3) a3 = ap[3];
            lA[arow][akx + 0] = a0; lA[arow][akx + 1] = a1;
            lA[arow][akx + 2] = a2; lA[arow][akx + 3] = a3;
        }
        {
            float4 bv0 = make_float4(0.f, 0.f, 0.f, 0.f);
            float4 bv1 = make_float4(0.f, 0.f, 0.f, 0.f);
            if (k0 + bk < K) {
                const float4* bp = (const float4*)(Bm + (size_t)(k0 + bk) * 64 + bn);
                bv0 = bp[0]; bv1 = bp[1];
            }
            lB[bk][bn + 0] = bv0.x; lB[bk][bn + 1] = bv0.y;
            lB[bk][bn + 2] = bv0.z; lB[bk][bn + 3] = bv0.w;
            lB[bk][bn + 4] = bv1.x; lB[bk][bn + 5] = bv1.y;
            lB[bk][bn + 6] = bv1.z; lB[bk][bn + 7] = bv1.w;
        }
        __syncthreads();

        v2f afr[8], bfr[8];
        #pragma unroll
        for (int s = 0; s < 8; ++s) {
            const int kb = (s << 2) + (half << 1);
            afr[s].x = lA[arowf][kb];
            afr[s].y = lA[arowf][kb + 1];
            bfr[s].x = lB[kb][n0w + l16];
            bfr[s].y = lB[kb + 1][n0w + l16];
        }
        #pragma unroll
        for (int s = 0; s < 8; ++s) {
            acc = __builtin_amdgcn_wmma_f32_16x16x4_f32(
                      false, afr[s], false, bfr[s], (short)0, acc, false, false);
        }
        __syncthreads();
    }

    // ---- epilogue: C/D layout — VGPR r holds M=r (lanes 0-15) / M=r+8 ----
    const int col = n0w + l16;
    const float bvv = bias[col];
    #pragma unroll
    for (int r = 0; r < 8; ++r) {
        const int row = m0 + mtile * 16 + r + (half << 3);
        if (row < M) {
            float v = acc[r] + bvv;
            if (ACT) v = gelu_exact(v);
            D[(size_t)row * 64 + col] = v;
        }
    }
}

// =====================================================================
// Edge attention-weight kernel. One wave per edge.
// Lanes 0-15 carry sx (m[src]), lanes 16-31 carry dx (m[dst]),
// 4 contiguous floats per lane -> feature index f = 4*lane + j.
// =====================================================================
__global__ __launch_bounds__(256) void edge_w_kernel(
    const float* __restrict__ mfeat, const int* __restrict__ ei,
    const float* __restrict__ g1, const float* __restrict__ b1,
    const float* __restrict__ Wa, const float* __restrict__ ba,
    const float* __restrict__ g2, const float* __restrict__ b2,
    const float* __restrict__ Wb, const float* __restrict__ bbp,
    float* __restrict__ wbuf, float* __restrict__ wsum)
{
    const int lane = threadIdx.x & 31;
    const int e = blockIdx.x * 8 + (threadIdx.x >> 5);
    if (e >= ETOT) return;

    const int src  = ei[e];
    const int dst  = ei[ETOT + e];
    const int node = (lane < 16) ? src : dst;
    const int fb   = lane * 4;

    const float4 v = *(const float4*)(mfeat + (size_t)node * 64 + (size_t)(lane & 15) * 4);

    // LayerNorm over 128 features (wave reduction)
    float s1 = v.x + v.y + v.z + v.w;
    float s2 = v.x * v.x + v.y * v.y + v.z * v.z + v.w * v.w;
    #pragma unroll
    for (int off = 16; off; off >>= 1) {
        s1 += __shfl_xor(s1, off, 32);
        s2 += __shfl_xor(s2, off, 32);
    }
    const float mean = s1 * (1.0f / 128.0f);
    const float inv  = rsqrtf(s2 * (1.0f / 128.0f) - mean * mean + 1e-5f);

    const float n0 = (v.x - mean) * inv * g1[fb + 0] + b1[fb + 0];
    const float n1 = (v.y - mean) * inv * g1[fb + 1] + b1[fb + 1];
    const float n2 = (v.z - mean) * inv * g1[fb + 2] + b1[fb + 2];
    const float n3 = (v.w - mean) * inv * g1[fb + 3] + b1[fb + 3];

    // [128] @ Wa[128,4]: per-lane partials, 4 wave reductions
    float p0 = n0 * Wa[(fb+0)*4+0] + n1 * Wa[(fb+1)*4+0] + n2 * Wa[(fb+2)*4+0] + n3 * Wa[(fb+3)*4+0];
    float p1 = n0 * Wa[(fb+0)*4+1] + n1 * Wa[(fb+1)*4+1] + n2 * Wa[(fb+2)*4+1] + n3 * Wa[(fb+3)*4+1];
    float p2 = n0 * Wa[(fb+0)*4+2] + n1 * Wa[(fb+1)*4+2] + n2 * Wa[(fb+2)*4+2] + n3 * Wa[(fb+3)*4+2];
    float p3 = n0 * Wa[(fb+0)*4+3] + n1 * Wa[(fb+1)*4+3] + n2 * Wa[(fb+2)*4+3] + n3 * Wa[(fb+3)*4+3];
    #pragma unroll
    for (int off = 16; off; off >>= 1) {
        p0 += __shfl_xor(p0, off, 32);
        p1 += __shfl_xor(p1, off, 32);
        p2 += __shfl_xor(p2, off, 32);
        p3 += __shfl_xor(p3, off, 32);
    }

    const float q0 = gelu_exact(p0 + ba[0]);
    const float q1 = gelu_exact(p1 + ba[1]);
    const float q2 = gelu_exact(p2 + ba[2]);
    const float q3 = gelu_exact(p3 + ba[3]);

    // LayerNorm over 4 (all lanes hold all 4 values)
    const float mu  = 0.25f * (q0 + q1 + q2 + q3);
    const float va  = 0.25f * (q0*q0 + q1*q1 + q2*q2 + q3*q3) - mu * mu;
    const float iv2 = rsqrtf(va + 1e-5f);
    const float t0 = (q0 - mu) * iv2 * g2[0] + b2[0];
    const float t1 = (q1 - mu) * iv2 * g2[1] + b2[1];
    const float t2 = (q2 - mu) * iv2 * g2[2] + b2[2];
    const float t3 = (q3 - mu) * iv2 * g2[3] + b2[3];

    const float sc = t0 * Wb[0] + t1 * Wb[1] + t2 * Wb[2] + t3 * Wb[3] + bbp[0];
    const float w  = expf(sc);

    if (lane == 0) {
        wbuf[e] = w;
        unsafeAtomicAdd(&wsum[dst], w);
    }
}

// =====================================================================
// Edge aggregation: agg[dst] += m[src] * (w / max(wsum[dst],1e-8))
// One wave per edge; 2 floats/lane; native f32 global atomics (L2-resident).
// =====================================================================
__global__ __launch_bounds__(256) void edge_agg_kernel(
    const float* __restrict__ mfeat, const int* __restrict__ ei,
    const float* __restrict__ wbuf, const float* __restrict__ wsum,
    float* __restrict__ agg)
{
    const int lane = threadIdx.x & 31;
    const int e = blockIdx.x * 8 + (threadIdx.x >> 5);
    if (e >= ETOT) return;

    const int src = ei[e];
    const int dst = ei[ETOT + e];
    const float nw = wbuf[e] / fmaxf(wsum[dst], 1e-8f);

    const float2 sv = *(const float2*)(mfeat + (size_t)src * 64 + lane * 2);
    float* ap = agg + (size_t)dst * 64 + lane * 2;
    unsafeAtomicAdd(ap,     sv.x * nw);
    unsafeAtomicAdd(ap + 1, sv.y * nw);
}

// =====================================================================
// Head: per graph, scores over 2048 nodes (node 0 dropped), softmax,
// weighted sums of exc/vxc. One 256-thread block per graph.
// =====================================================================
__global__ __launch_bounds__(256) void head_kernel(
    const float* __restrict__ h, const float* __restrict__ exc,
    const float* __restrict__ vxc, const float* __restrict__ Wh,
    const float* __restrict__ bh, float* __restrict__ out)
{
    __shared__ float red[256];
    __shared__ float sWh[64];
    const int g = blockIdx.x;
    const int tid = threadIdx.x;
    if (tid < 64) sWh[tid] = Wh[tid];
    __syncthreads();

    const float bh0 = bh[0];
    const float* hb = h + (size_t)g * NPG * 64;

    float sc[8];
    float lmax = -3.4e38f;
    #pragma unroll
    for (int i = 0; i < 8; ++i) {
        const int node = 1 + tid + i * 256;
        const float* hv = hb + (size_t)node * 64;
        float s = bh0;
        #pragma unroll
        for (int c = 0; c < 64; ++c) s += hv[c] * sWh[c];
        sc[i] = s;
        lmax = fmaxf(lmax, s);
    }

    red[tid] = lmax; __syncthreads();
    for (int o = 128; o; o >>= 1) {
        if (tid < o) red[tid] = fmaxf(red[tid], red[tid + o]);
        __syncthreads();
    }
    const float gmax = red[0]; __syncthreads();

    float se = 0.f, sexc = 0.f, svxc = 0.f;
    const float* eb = exc + (size_t)g * NPG;
    const float* vb = vxc + (size_t)g * NPG;
    #pragma unroll
    for (int i = 0; i < 8; ++i) {
        const int node = 1 + tid + i * 256;
        const float p = expf(sc[i] - gmax);
        se += p; sexc += p * eb[node]; svxc += p * vb[node];
    }

    red[tid] = se; __syncthreads();
    for (int o = 128; o; o >>= 1) { if (tid < o) red[tid] += red[tid + o]; __syncthreads(); }
    const float gse = red[0]; __syncthreads();

    red[tid] = sexc; __syncthreads();
    for (int o = 128; o; o >>= 1) { if (tid < o) red[tid] += red[tid + o]; __syncthreads(); }
    const float gsexc = red[0]; __syncthreads();

    red[tid] = svxc; __syncthreads();
    for (int o = 128; o; o >>= 1) { if (tid < o) red[tid] += red[tid + o]; __syncthreads(); }
    const float gsvxc = red[0];

    if (tid == 0) {
        out[g]       = gsexc / gse;  // pred_exc
        out[BGR + g] = gsvxc / gse;  // pred_vxc
    }
}

__global__ void zero_f32(float* __restrict__ p, int n)
{
    const int i = blockIdx.x * 256 + threadIdx.x;
    if (i < n) p[i] = 0.f;
}

// =====================================================================
extern "C" void kernel_launch(void* const* d_in, const int* in_sizes, int n_in,
                              void* d_out, int out_size, void* d_ws, size_t ws_size,
                              hipStream_t stream)
{
    const float* x    = (const float*)d_in[0];
    const float* exc  = (const float*)d_in[1];
    const float* vxc  = (const float*)d_in[2];
    const int*   ei   = (const int*)  d_in[3];
    const float* W_in = (const float*)d_in[4];
    const float* b_in = (const float*)d_in[5];
    const float* Wd1  = (const float*)d_in[6];
    const float* bd1  = (const float*)d_in[7];
    const float* Wd2  = (const float*)d_in[8];
    const float* bd2  = (const float*)d_in[9];
    const float* ln1g = (const float*)d_in[10];
    const float* ln1b = (const float*)d_in[11];
    const float* Wa   = (const float*)d_in[12];
    const float* ba   = (const float*)d_in[13];
    const float* ln2g = (const float*)d_in[14];
    const float* ln2b = (const float*)d_in[15];
    const float* Wb   = (const float*)d_in[16];
    const float* bb   = (const float*)d_in[17];
    const float* Wh   = (const float*)d_in[18];
    const float* bh   = (const float*)d_in[19];
    float* out = (float*)d_out;

    // workspace layout (f32): h | m | agg | wsum | wbuf  (~27.4 MB)
    float* h    = (float*)d_ws;
    float* m    = h    + (size_t)NTOT * 64;
    float* agg  = m    + (size_t)NTOT * 64;
    float* wsum = agg  + (size_t)NTOT * 64;
    float* wbuf = wsum + (size_t)NTOT;

    const int gemm_grid = (NTOT + 31) / 32;                 // 1025
    const int egrid     = (ETOT + 7) / 8;                   // 65568
    const int zcount    = NTOT * 64 + NTOT;                 // agg + wsum (contiguous)
    const int zgrid     = (zcount + 255) / 256;

    // input projection: h = x @ W_in + b_in
    gemm64_wmma<false><<<gemm_grid, 256, 0, stream>>>(x, W_in, b_in, h, NTOT, FINC);

    for (int l = 0; l < LNUM; ++l) {
        // m = gelu(h @ Wd1[l] + bd1[l])
        gemm64_wmma<true><<<gemm_grid, 256, 0, stream>>>(h, Wd1 + (size_t)l * 64 * 64,
                                                         bd1 + l * 64, m, NTOT, 64);
        zero_f32<<<zgrid, 256, 0, stream>>>(agg, zcount);
        edge_w_kernel<<<egrid, 256, 0, stream>>>(m, ei,
            ln1g + l * 128, ln1b + l * 128,
            Wa + (size_t)l * 512, ba + l * 4,
            ln2g + l * 4, ln2b + l * 4,
            Wb + l * 4, bb + l, wbuf, wsum);
        edge_agg_kernel<<<egrid, 256, 0, stream>>>(m, ei, wbuf, wsum, agg);
        // h = gelu(agg @ Wd2[l] + bd2[l])
        gemm64_wmma<true><<<gemm_grid, 256, 0, stream>>>(agg, Wd2 + (size_t)l * 64 * 64,
                                                         bd2 + l * 64, h, NTOT, 64);
    }

    head_kernel<<<BGR, 256, 0, stream>>>(h, exc, vxc, Wh, bh, out);
}